// GGUp_1090921693816
// MI455X (gfx1250) — compile-verified
//
#include <hip/hip_runtime.h>
#include <hip/hip_bf16.h>
#include <stdint.h>

// GGUp (CARAFE-style 2x content-aware upsampler), MI455X / gfx1250.
// Memory-bound: ~335 MB min HBM traffic -> ~14.4 us at 23.3 TB/s.
// x (67 MB) fits in the 192 MB L2, so the two-pass mean read and tile
// halo re-reads are L2 hits. Output stream (268 MB) uses NT stores to
// avoid evicting x from L2. Tile staging uses CDNA5 async global->LDS
// loads (ASYNCcnt) with double buffering.
//
// NOTE: ggup_apply is defined first so the disasm snippet shows its body
// (async global->LDS staging + s_wait_asynccnt + ds loads).

#define NB 4
#define NC 256
#define NH 128
#define NW 128
#define HW (NH * NW)           // 16384
#define OW (2 * NW)            // 256
#define OHW (4 * HW)           // 65536 output elems per (b,c) plane
#define TILE 16
#define HALO (TILE + 2)        // 18
#define NSLOT (HALO * HALO)    // 324
#define NTHREADS 256
#define CPG 64                 // channels per workgroup
#define NCG (NC / CPG)         // 4

#ifndef __has_builtin
#define __has_builtin(x) 0
#endif
#if __has_builtin(__builtin_amdgcn_global_load_async_to_lds_b32) && \
    __has_builtin(__builtin_amdgcn_s_wait_asynccnt)
#define USE_ASYNC_LDS 1
#else
#define USE_ASYNC_LDS 0
#endif

#define AS1 __attribute__((address_space(1)))
#define AS3 __attribute__((address_space(3)))

typedef float v2f __attribute__((ext_vector_type(2)));
typedef float v4f __attribute__((ext_vector_type(4)));

// ---------------------------------------------------------------------------
// Pass 2: masks (softmax over 9 taps, 4 subpixels) + weighted gather.
// One block = 16x16 pixel tile x 64 channels. Thread = pixel; its 36 mask
// values live in VGPRs for the whole channel loop. x tiles (18x18 halo) are
// double-buffered in LDS via async global->LDS loads.
// ---------------------------------------------------------------------------
__global__ __launch_bounds__(NTHREADS) void ggup_apply(const float* __restrict__ x,
                                                       const float* __restrict__ gk, // (9,4) row-major
                                                       const float* __restrict__ mean,
                                                       float* __restrict__ out)
{
  __shared__ float sMean[NSLOT];
  __shared__ float sX[2][NSLOT];

  const int tid  = threadIdx.x;
  const int tile = blockIdx.x;          // 0..63 (8x8 tiles)
  const int cg   = blockIdx.y;          // channel group
  const int b    = blockIdx.z;
  const int h0   = (tile >> 3) * TILE;
  const int w0   = (tile & 7) * TILE;

  // Staging-slot geometry (fixed per workgroup). Each thread owns <=2 slots.
  const int  s0   = tid;
  const int  gy0  = h0 - 1 + s0 / HALO;
  const int  gx0  = w0 - 1 + s0 % HALO;
  const bool inb0 = ((unsigned)gy0 < NH) & ((unsigned)gx0 < NW);
  const int  off0 = inb0 ? (gy0 * NW + gx0) : 0;

  const int  s1   = tid + NTHREADS;
  const bool has1 = (s1 < NSLOT);
  bool inb1 = false; int off1 = 0;
  if (has1) {
    const int gy1 = h0 - 1 + s1 / HALO;
    const int gx1 = w0 - 1 + s1 % HALO;
    inb1 = ((unsigned)gy1 < NH) & ((unsigned)gx1 < NW);
    off1 = inb1 ? (gy1 * NW + gx1) : 0;
  }

  // Stage mean halo tile; pre-zero the (fixed) out-of-image x slots once.
  // Async loads are predicated off for OOB slots, so DS zeros never race
  // with async LDS writes.
  {
    const float* __restrict__ mb = mean + (size_t)b * HW;
    sMean[s0] = inb0 ? mb[off0] : 0.0f;
    if (!inb0) { sX[0][s0] = 0.0f; sX[1][s0] = 0.0f; }
    if (has1) {
      sMean[s1] = inb1 ? mb[off1] : 0.0f;
      if (!inb1) { sX[0][s1] = 0.0f; sX[1][s1] = 0.0f; }
    }
  }

  const size_t planeBase = (size_t)(b * NC + cg * CPG) * HW;

#if USE_ASYNC_LDS
#define STAGE(cc, buf)                                                         \
  do {                                                                         \
    const float* gp_ = x + planeBase + (size_t)(cc) * HW;                      \
    if (inb0)                                                                  \
      __builtin_amdgcn_global_load_async_to_lds_b32(                           \
          (AS1 int*)(gp_ + off0), (AS3 int*)&sX[buf][s0], 0, 0);               \
    if (has1 && inb1)                                                          \
      __builtin_amdgcn_global_load_async_to_lds_b32(                           \
          (AS1 int*)(gp_ + off1), (AS3 int*)&sX[buf][s1], 0, 0);               \
  } while (0)
#define WAIT_STAGE() __builtin_amdgcn_s_wait_asynccnt(0)
#else
#define STAGE(cc, buf)                                                         \
  do {                                                                         \
    const float* gp_ = x + planeBase + (size_t)(cc) * HW;                      \
    sX[buf][s0] = inb0 ? gp_[off0] : 0.0f;                                     \
    if (has1) sX[buf][s1] = inb1 ? gp_[off1] : 0.0f;                           \
  } while (0)
#define WAIT_STAGE()
#endif

  STAGE(0, 0);          // start channel 0 while we do the mask math
  __syncthreads();      // sMean + OOB zeros visible

  // Per-pixel masks, kept entirely in registers (36 VGPRs/thread).
  const int ty = tid >> 4;
  const int tx = tid & 15;
  float g[9];
  {
    const float mc = sMean[(ty + 1) * HALO + (tx + 1)];
    #pragma unroll
    for (int n = 0; n < 9; ++n) {
      const float d = sMean[(ty + n / 3) * HALO + (tx + n % 3)] - mc;
      g[n] = 1.0f / (d * d + 1.0f);
    }
  }
  float m[9][4];
  #pragma unroll
  for (int q = 0; q < 4; ++q) {
    float l[9];
    float mx = -3.402823e38f;
    #pragma unroll
    for (int n = 0; n < 9; ++n) { l[n] = g[n] * gk[n * 4 + q]; mx = fmaxf(mx, l[n]); }
    float s = 0.0f;
    #pragma unroll
    for (int n = 0; n < 9; ++n) { const float e = __expf(l[n] - mx); m[n][q] = e; s += e; }
    const float inv = 1.0f / s;
    #pragma unroll
    for (int n = 0; n < 9; ++n) m[n][q] *= inv;
  }

  WAIT_STAGE();
  __syncthreads();      // sX[0] ready for all waves

  float* __restrict__ obase =
      out + (size_t)(b * NC + cg * CPG) * OHW
          + (size_t)(2 * (h0 + ty)) * OW + (size_t)(2 * (w0 + tx));

  for (int c = 0; c < CPG; ++c) {
    if (c + 1 < CPG) STAGE(c + 1, (c + 1) & 1);   // prefetch next channel

    const float* __restrict__ xb = sX[c & 1];
    float o0 = 0.f, o1 = 0.f, o2 = 0.f, o3 = 0.f;
    #pragma unroll
    for (int n = 0; n < 9; ++n) {
      const float xv = xb[(ty + n / 3) * HALO + (tx + n % 3)];
      o0 += m[n][0] * xv; o1 += m[n][1] * xv;
      o2 += m[n][2] * xv; o3 += m[n][3] * xv;
    }
    // out[2h+ry, 2w+rx], q = ry*2+rx; two coalesced 8B NT stores.
    float* orow = obase + (size_t)c * OHW;
    v2f r0; r0.x = o0; r0.y = o1;
    v2f r1; r1.x = o2; r1.y = o3;
    __builtin_nontemporal_store(r0, (v2f*)orow);
    __builtin_nontemporal_store(r1, (v2f*)(orow + OW));

    WAIT_STAGE();
    __syncthreads();    // buffer (c+1) ready; buffer (c) free for reuse
  }
}

// ---------------------------------------------------------------------------
// Pass 1: channel mean, mean[b,h,w] = (1/256) * sum_c x[b,c,h,w].
// float4-vectorized, fully coalesced; also warms L2 with all of x.
// ---------------------------------------------------------------------------
__global__ __launch_bounds__(NTHREADS) void ggup_mean(const float* __restrict__ x,
                                                      float* __restrict__ mean)
{
  const int t = blockIdx.x * NTHREADS + threadIdx.x;  // 0 .. NB*HW/4-1
  const int b = t >> 12;                              // HW/4 = 4096 vec4/plane
  const int r = t & 4095;
  const v4f* __restrict__ p = (const v4f*)x + (size_t)b * NC * (HW / 4) + r;
  v4f acc = (v4f)0.0f;
  #pragma unroll 8
  for (int c = 0; c < NC; ++c)
    acc += p[(size_t)c * (HW / 4)];
  ((v4f*)mean)[t] = acc * (1.0f / NC);
}

extern "C" void kernel_launch(void* const* d_in, const int* in_sizes, int n_in,
                              void* d_out, int out_size, void* d_ws, size_t ws_size,
                              hipStream_t stream) {
  const float* x  = (const float*)d_in[0];   // (4,256,128,128) f32
  const float* gk = (const float*)d_in[1];   // (9,4) f32
  float* mean = (float*)d_ws;                // NB*HW floats = 256 KB scratch
  float* out  = (float*)d_out;               // (4,256,256,256) f32

  ggup_mean<<<(NB * HW / 4) / NTHREADS, NTHREADS, 0, stream>>>(x, mean);

  dim3 grid(64, NCG, NB);                    // 1024 blocks x 8 waves
  ggup_apply<<<grid, NTHREADS, 0, stream>>>(x, gk, mean, out);
}